// AttentivePooler_26190710571236
// MI455X (gfx1250) — compile-verified
//
#include <hip/hip_runtime.h>
#include <math.h>
#include <stdint.h>

// AttentivePooler fused kernel for gfx1250 (MI455X).
// Memory-bound: 256 MiB fp32 embeddings streamed exactly once (~11us @ 23.3TB/s).
// - Scores via full-precision V_WMMA_F32_16X16X4_F32 (reference is fp32; no downcast).
// - Double-buffered GLOBAL_LOAD_ASYNC_TO_LDS_B128 (ASYNCcnt) overlaps HBM with compute.

typedef __attribute__((ext_vector_type(2))) float v2f;
typedef __attribute__((ext_vector_type(4))) float v4f;
typedef __attribute__((ext_vector_type(8))) float v8f;

#define BB 64
#define TT 4096
#define HH 256
#define CHUNKS 16
#define TC (TT / CHUNKS)     // 256 tokens per block
#define TILE 16
#define NTILES (TC / TILE)   // 16 tiles per block
#define LDSS 260             // padded LDS row stride (floats): rows hit distinct banks

__global__ __launch_bounds__(256) void attnpool_pass1(
    const float* __restrict__ emb, const float* __restrict__ mask,
    const float* __restrict__ query, float* __restrict__ ws)
{
    __shared__ __attribute__((aligned(16))) float lds_E[2][TILE * LDSS]; // 2 x 16.25KB
    __shared__ float lds_q[HH];
    __shared__ float lds_partial[8][TILE];
    __shared__ float lds_mask[TILE];
    __shared__ float lds_wt[TILE];
    __shared__ float lds_r[1];

    const int tid   = threadIdx.x;
    const int wave  = tid >> 5;
    const int lane  = tid & 31;
    const int b     = blockIdx.x / CHUNKS;
    const int chunk = blockIdx.x % CHUNKS;
    const int t0base = chunk * TC;
    const int h = tid;                    // each thread owns one hidden index

    lds_q[tid] = query[tid];              // stage query once

    float m = -INFINITY;                  // online-softmax state (wave 0 authoritative)
    float s = 0.f;
    float acc = 0.f;                      // running weighted sum for hidden h

    const size_t rowbase = ((size_t)b * TT + t0base) * HH;

    // Async prefetch of one 16x256 fp32 tile into LDS buffer `buf`.
    // Per wave: 4 x global_load_async_to_lds_b128 (32 lanes x 16B each = 2KB).
    auto prefetch = [&](int buf, int tile) {
        const v4f* gsrc = (const v4f*)(emb + rowbase + (size_t)tile * TILE * HH);
#pragma unroll
        for (int i = 0; i < 4; ++i) {
            int idx = tid + i * 256;              // [0,1024) float4s
            int t   = idx >> 6;                   // token row 0..15
            int c4  = idx & 63;                   // float4 column
            const v4f* src = &gsrc[(size_t)t * (HH / 4) + c4];
            uint32_t dst = (uint32_t)(uintptr_t)&lds_E[buf][t * LDSS + c4 * 4];
            asm volatile("global_load_async_to_lds_b128 %0, %1, off"
                         :: "v"(dst), "v"(src) : "memory");
        }
    };

    prefetch(0, 0);

    for (int tile = 0; tile < NTILES; ++tile) {
        const int cur = tile & 1;

        // Kick off next tile into the other buffer (safe: last iteration ended
        // with a barrier, so no wave is still reading buffer cur^1).
        if (tile + 1 < NTILES) prefetch(cur ^ 1, tile + 1);

        if (tid < TILE)
            lds_mask[tid] = mask[(size_t)b * TT + t0base + tile * TILE + tid];

        // Async loads complete in order: <=4 outstanding => current tile done.
        if (tile + 1 < NTILES) {
            asm volatile("s_wait_asynccnt 0x4" ::: "memory");
        } else {
            asm volatile("s_wait_asynccnt 0x0" ::: "memory");
        }
        __syncthreads();   // bar A: tile `cur` visible to all waves

        // ---- scores via fp32 WMMA: wave w covers H-range [w*32, w*32+32) ----
        {
            v8f c = {0.f, 0.f, 0.f, 0.f, 0.f, 0.f, 0.f, 0.f};
            const float* rowp = &lds_E[cur][(lane & 15) * LDSS]; // A row = token lane%16
            const int ksel = (lane >> 4) << 1;                   // upper half-lanes: K+2
#pragma unroll
            for (int i = 0; i < 8; ++i) {
                int kk = wave * 32 + i * 4 + ksel;
                v2f a;  a.x  = rowp[kk];   a.y  = rowp[kk + 1];  // A: 16x4 E slice
                v2f bq; bq.x = lds_q[kk];  bq.y = lds_q[kk + 1]; // B: q bcast to 16 cols
                c = __builtin_amdgcn_wmma_f32_16x16x4_f32(
                        false, a, false, bq, (short)0, c, false, false);
            }
            // D columns identical: lane0 holds tokens 0..7, lane16 tokens 8..15
            if (lane == 0 || lane == 16) {
                int base = (lane == 0) ? 0 : 8;
#pragma unroll
                for (int r = 0; r < 8; ++r)
                    lds_partial[wave][base + r] = c[r];
            }
        }
        __syncthreads();   // bar B: partial scores ready

        // ---- wave 0: reduce partials + online-softmax bookkeeping (once) ----
        if (wave == 0) {
            int t = lane & 15;                    // lanes 16-31 mirror 0-15
            float v = 0.f;
#pragma unroll
            for (int w = 0; w < 8; ++w) v += lds_partial[w][t];
            v = (lds_mask[t] == 0.f) ? -INFINITY : v;
            float mt = v;
#pragma unroll
            for (int o = 8; o >= 1; o >>= 1)
                mt = fmaxf(mt, __shfl_xor(mt, o, 32));
            float newm = fmaxf(m, mt);
            float r  = (m == -INFINITY) ? 0.f : __expf(m - newm);
            float wv = (v == -INFINITY) ? 0.f : __expf(v - newm);
            float wsum = wv;
#pragma unroll
            for (int o = 8; o >= 1; o >>= 1)
                wsum += __shfl_xor(wsum, o, 32);
            s = s * r + wsum;
            m = newm;
            if (lane < 16) lds_wt[lane] = wv;
            if (lane == 0) lds_r[0] = r;
        }
        __syncthreads();   // bar C: weights ready

        // ---- all 256 threads: rescale + accumulate weighted tile into acc[h] ----
        {
            float a2 = acc * lds_r[0];
#pragma unroll
            for (int t = 0; t < TILE; ++t)
                a2 = fmaf(lds_wt[t], lds_E[cur][t * LDSS + h], a2);
            acc = a2;
        }
        __syncthreads();   // bar D: buffer `cur` free for async overwrite next round
    }

    // partial record per block: [0]=m, [1]=s, [2+h]=acc[h]
    float* outp = ws + (size_t)blockIdx.x * (HH + 2);
    if (tid == 0) { outp[0] = m; outp[1] = s; }
    outp[2 + h] = acc;
}

__global__ __launch_bounds__(256) void attnpool_pass2(
    const float* __restrict__ ws, float* __restrict__ out)
{
    const int b = blockIdx.x;
    const int h = threadIdx.x;
    float M = -INFINITY;
#pragma unroll
    for (int k = 0; k < CHUNKS; ++k)
        M = fmaxf(M, ws[(size_t)(b * CHUNKS + k) * (HH + 2)]);
    float S = 0.f, P = 0.f;
#pragma unroll
    for (int k = 0; k < CHUNKS; ++k) {
        const float* p = ws + (size_t)(b * CHUNKS + k) * (HH + 2);
        float mk = p[0];
        float e = (mk == -INFINITY) ? 0.f : __expf(mk - M);
        S += p[1] * e;
        P += p[2 + h] * e;
    }
    out[(size_t)b * HH + h] = P / S;
}

extern "C" void kernel_launch(void* const* d_in, const int* in_sizes, int n_in,
                              void* d_out, int out_size, void* d_ws, size_t ws_size,
                              hipStream_t stream) {
    (void)in_sizes; (void)n_in; (void)out_size; (void)ws_size;
    const float* emb   = (const float*)d_in[0];
    const float* maskp = (const float*)d_in[1];
    const float* query = (const float*)d_in[2];
    float* out = (float*)d_out;
    float* ws  = (float*)d_ws;   // needs 1024*258*4 B = 1.03 MB

    attnpool_pass1<<<BB * CHUNKS, 256, 0, stream>>>(emb, maskp, query, ws);
    attnpool_pass2<<<BB, 256, 0, stream>>>(ws, out);
}